// FlexibleAttention_15272903705025
// MI455X (gfx1250) — compile-verified
//
#include <hip/hip_runtime.h>
#include <math.h>

// FlexibleAttention, N=1024, DIM=128, fp32.
// Algebraic fusion: out_i = (Σ_j a_ij V_j) + (Σ_j a_ij relu(Q_i+K_j)) @ W_Ev + b_Ev
// => never materialize the [n,n,d] edge tensor; one-pass online softmax per row.

#define NPTS 1024
#define DIM  128

typedef float v2f __attribute__((ext_vector_type(2)));
typedef float v8f __attribute__((ext_vector_type(8)));

// ---------------------------------------------------------------------------
// Kernel 1: Q/K/V = x @ W + b   via V_WMMA_F32_16X16X4_F32
// grid = (NPTS/16, 3), block = 256 (8 waves; wave w owns n-tile w*16)
// blockIdx.y selects which of Q/K/V this block computes.
// Fragment layout (ISA 7.12.2, 32-bit):
//   A (16x4): lane L holds M=L%16, regs {K=2*(L/16), K=2*(L/16)+1}
//   B (4x16): lane L holds N=L%16, regs {K=2*(L/16), K=2*(L/16)+1}
//   D (16x16): c[r] = D[8*(L/16)+r][L%16]
// ---------------------------------------------------------------------------
__global__ __launch_bounds__(256) void qkv_gemm_wmma(
    const float* __restrict__ x,
    const float* __restrict__ Wq, const float* __restrict__ bq,
    const float* __restrict__ Wk, const float* __restrict__ bk,
    const float* __restrict__ Wv, const float* __restrict__ bv,
    float* __restrict__ Q, float* __restrict__ K, float* __restrict__ V)
{
    const int lane = threadIdx.x & 31;
    const int wave = threadIdx.x >> 5;
    const int lo   = lane & 15;
    const int hi   = lane >> 4;
    const int i0   = blockIdx.x * 16;
    const int n0   = wave * 16;

    const float* W; const float* bias; float* out;
    if (blockIdx.y == 0)      { W = Wq; bias = bq; out = Q; }
    else if (blockIdx.y == 1) { W = Wk; bias = bk; out = K; }
    else                      { W = Wv; bias = bv; out = V; }

    v8f c = {0.f,0.f,0.f,0.f,0.f,0.f,0.f,0.f};
    const float* xrow = x + (size_t)(i0 + lo) * DIM;

#pragma unroll
    for (int k0 = 0; k0 < DIM; k0 += 4) {
        const int ka = k0 + 2 * hi;
        v2f a; a.x = xrow[ka];
               a.y = xrow[ka + 1];
        v2f b; b.x = W[(size_t)ka       * DIM + n0 + lo];
               b.y = W[(size_t)(ka + 1) * DIM + n0 + lo];
        c = __builtin_amdgcn_wmma_f32_16x16x4_f32(
                false, a, false, b, (short)0, c, false, false);
    }

    const float bn = bias[n0 + lo];
#pragma unroll
    for (int r = 0; r < 8; ++r) {
        const int m = hi * 8 + r;
        out[(size_t)(i0 + m) * DIM + n0 + lo] = c[r] + bn;
    }
}

// ---------------------------------------------------------------------------
// Kernel 2: per-row online softmax over j, accumulating
//   S_i  = (Σ_j p_ij relu(Q_i+K_j)) / Σ_j p_ij
//   AV_i = (Σ_j p_ij V_j)           / Σ_j p_ij
// grid = NPTS/8 blocks, block = 256 threads = 8 waves; wave w owns row i.
// K/V staged through LDS in 32-row tiles, shared by the 8 waves (8x reuse).
// Each lane holds 4 dims (float4): dims [4*lane, 4*lane+3].
// ---------------------------------------------------------------------------
__global__ __launch_bounds__(256) void edge_softmax_pass(
    const float* __restrict__ Q, const float* __restrict__ K,
    const float* __restrict__ V,
    const float* __restrict__ wA, const float* __restrict__ bA,
    float* __restrict__ S, float* __restrict__ AV)
{
    __shared__ float Kt[32 * DIM];   // 16 KB
    __shared__ float Vt[32 * DIM];   // 16 KB

    const int tid  = threadIdx.x;
    const int lane = tid & 31;
    const int wave = tid >> 5;
    const int i    = blockIdx.x * 8 + wave;

    const float4 q4  = *(const float4*)(Q  + (size_t)i * DIM + lane * 4);
    const float4 wa4 = *(const float4*)(wA + lane * 4);
    const float  ba  = bA[0];

    float  mrun  = -1e30f;
    float  denom = 0.f;
    float4 s4    = {0.f,0.f,0.f,0.f};
    float4 av4   = {0.f,0.f,0.f,0.f};

    for (int j0 = 0; j0 < NPTS; j0 += 32) {
        __syncthreads();
        // Cooperative stage of 32 rows of K and V (4096 floats each).
        const float4* Kg = (const float4*)(K + (size_t)j0 * DIM);
        const float4* Vg = (const float4*)(V + (size_t)j0 * DIM);
        float4* Ks = (float4*)Kt;
        float4* Vs = (float4*)Vt;
#pragma unroll
        for (int p = 0; p < 4; ++p) {
            const int f = tid + p * 256;   // 1024 float4 per tile
            Ks[f] = Kg[f];
            Vs[f] = Vg[f];
        }
        __syncthreads();

#pragma unroll 4
        for (int jj = 0; jj < 32; ++jj) {
            const float4 k4 = *(const float4*)(Kt + jj * DIM + lane * 4);
            float4 r;
            r.x = fmaxf(q4.x + k4.x, 0.f);
            r.y = fmaxf(q4.y + k4.y, 0.f);
            r.z = fmaxf(q4.z + k4.z, 0.f);
            r.w = fmaxf(q4.w + k4.w, 0.f);

            // e_ij = relu(Q_i+K_j) . W_A + b_A  (wave32 butterfly reduce)
            float part = r.x*wa4.x + r.y*wa4.y + r.z*wa4.z + r.w*wa4.w;
#pragma unroll
            for (int off = 16; off > 0; off >>= 1)
                part += __shfl_xor(part, off, 32);
            const float e = part + ba;

            // Online softmax update.
            const float mnew  = fmaxf(mrun, e);
            const float scale = __expf(mrun - mnew);
            const float pj    = __expf(e - mnew);
            denom = denom * scale + pj;

            const float4 v4 = *(const float4*)(Vt + jj * DIM + lane * 4);
            s4.x  = s4.x  * scale + pj * r.x;
            s4.y  = s4.y  * scale + pj * r.y;
            s4.z  = s4.z  * scale + pj * r.z;
            s4.w  = s4.w  * scale + pj * r.w;
            av4.x = av4.x * scale + pj * v4.x;
            av4.y = av4.y * scale + pj * v4.y;
            av4.z = av4.z * scale + pj * v4.z;
            av4.w = av4.w * scale + pj * v4.w;
            mrun = mnew;
        }
    }

    const float inv = 1.0f / denom;
    float4 so = { s4.x*inv,  s4.y*inv,  s4.z*inv,  s4.w*inv };
    float4 ao = { av4.x*inv, av4.y*inv, av4.z*inv, av4.w*inv };
    *(float4*)(S  + (size_t)i * DIM + lane * 4) = so;
    *(float4*)(AV + (size_t)i * DIM + lane * 4) = ao;
}

// ---------------------------------------------------------------------------
// Kernel 3: out = AV + S @ W_Ev + b_Ev   (WMMA f32 16x16x4, same tiling as K1)
// grid = NPTS/16 blocks, block = 256 (wave w owns n-tile w*16)
// ---------------------------------------------------------------------------
__global__ __launch_bounds__(256) void out_gemm_wmma(
    const float* __restrict__ S, const float* __restrict__ AV,
    const float* __restrict__ Wev, const float* __restrict__ bev,
    float* __restrict__ out)
{
    const int lane = threadIdx.x & 31;
    const int wave = threadIdx.x >> 5;
    const int lo   = lane & 15;
    const int hi   = lane >> 4;
    const int i0   = blockIdx.x * 16;
    const int n0   = wave * 16;

    v8f c = {0.f,0.f,0.f,0.f,0.f,0.f,0.f,0.f};
    const float* srow = S + (size_t)(i0 + lo) * DIM;

#pragma unroll
    for (int k0 = 0; k0 < DIM; k0 += 4) {
        const int ka = k0 + 2 * hi;
        v2f a; a.x = srow[ka];
               a.y = srow[ka + 1];
        v2f b; b.x = Wev[(size_t)ka       * DIM + n0 + lo];
               b.y = Wev[(size_t)(ka + 1) * DIM + n0 + lo];
        c = __builtin_amdgcn_wmma_f32_16x16x4_f32(
                false, a, false, b, (short)0, c, false, false);
    }

    const float bn = bev[n0 + lo];
#pragma unroll
    for (int r = 0; r < 8; ++r) {
        const int m = hi * 8 + r;
        const size_t idx = (size_t)(i0 + m) * DIM + n0 + lo;
        out[idx] = c[r] + AV[idx] + bn;
    }
}

// ---------------------------------------------------------------------------
// Launch: inputs in setup_inputs() order:
// 0:x 1:W_Q 2:b_Q 3:W_K 4:b_K 5:W_V 6:b_V 7:W_Ev 8:b_Ev 9:W_A 10:b_A
// Workspace: Q,K,V,S,AV = 5 * 1024*128 floats = 2.62 MB.
// ---------------------------------------------------------------------------
extern "C" void kernel_launch(void* const* d_in, const int* in_sizes, int n_in,
                              void* d_out, int out_size, void* d_ws, size_t ws_size,
                              hipStream_t stream) {
    const float* x   = (const float*)d_in[0];
    const float* Wq  = (const float*)d_in[1];
    const float* bq  = (const float*)d_in[2];
    const float* Wk  = (const float*)d_in[3];
    const float* bk  = (const float*)d_in[4];
    const float* Wv  = (const float*)d_in[5];
    const float* bv  = (const float*)d_in[6];
    const float* Wev = (const float*)d_in[7];
    const float* bev = (const float*)d_in[8];
    const float* wA  = (const float*)d_in[9];
    const float* bA  = (const float*)d_in[10];

    float* ws = (float*)d_ws;
    float* Q  = ws;
    float* K  = ws + 1 * NPTS * DIM;
    float* V  = ws + 2 * NPTS * DIM;
    float* S  = ws + 3 * NPTS * DIM;
    float* AV = ws + 4 * NPTS * DIM;

    qkv_gemm_wmma<<<dim3(NPTS / 16, 3), 256, 0, stream>>>(
        x, Wq, bq, Wk, bk, Wv, bv, Q, K, V);

    edge_softmax_pass<<<NPTS / 8, 256, 0, stream>>>(
        Q, K, V, wA, bA, S, AV);

    out_gemm_wmma<<<NPTS / 16, 256, 0, stream>>>(
        S, AV, Wev, bev, (float*)d_out);
}